// GAT_67508295958888
// MI455X (gfx1250) — compile-verified
//
#include <hip/hip_runtime.h>

#define N_NODES   100000
#define DEG       16
#define KDIM      256
#define NEG_SLOPE 0.2f
#define OUTD      41
#define L2W       164   // 4 heads * 41 classes
#define L2WP      176   // padded to multiple of 16

typedef float v2f __attribute__((ext_vector_type(2)));
typedef float v8f __attribute__((ext_vector_type(8)));

// -------------------------------------------------------------------------
// Pack W[256, ncols_src] into WMMA B-operand order, zero-padded to NTILES*16:
//   Bp[((kt*ntiles + tile)*32 + lane)*2 + v] =
//       W[(4*kt + 2*(lane>>4) + v) * ncols_src + 16*tile + (lane&15)]
// One thread per (kt, tile, lane) writes a float2.
// -------------------------------------------------------------------------
__global__ void gat_packW(const float* __restrict__ W, float* __restrict__ Bp,
                          int ncols_src, int ntiles) {
  const int t = blockIdx.x * blockDim.x + threadIdx.x;
  const int total = (KDIM / 4) * ntiles * 32;
  if (t >= total) return;
  const int lane = t & 31;
  const int tile = (t >> 5) % ntiles;
  const int kt   = (t >> 5) / ntiles;
  const int col  = tile * 16 + (lane & 15);
  const int krow = 4 * kt + 2 * (lane >> 4);
  float2 v;
  const bool ok = (col < ncols_src);
  v.x = ok ? W[(size_t)krow * ncols_src + col]       : 0.0f;
  v.y = ok ? W[(size_t)(krow + 1) * ncols_src + col] : 0.0f;
  *(float2*)(Bp + (size_t)t * 2) = v;
}

// -------------------------------------------------------------------------
// GEMM: C[N, NCOLS] = A[N, 256] @ W[256, NCOLS]  (fp32 WMMA 16x16x4)
// B pre-packed (see gat_packW) -> single coalesced b64 load per K-step,
// no guards, compile-time strides. One wave per 16x16 tile.
// -------------------------------------------------------------------------
template <int NCOLS>
__global__ void gat_gemm(const float* __restrict__ A, const float* __restrict__ Bp,
                         float* __restrict__ C) {
  constexpr int NTILES = NCOLS / 16;
  const int lane = threadIdx.x & 31;
  const int wave = threadIdx.x >> 5;
  const int hf   = lane >> 4;
  const int lm   = lane & 15;
  const int rowbase = blockIdx.x * 16;
  const int tile    = blockIdx.y * 8 + wave;
  if (tile >= NTILES) return;                 // wave-uniform exit
  const int col = tile * 16 + lm;

  // A element for this lane: row = rowbase+lm, K = 4*kt + 2*hf + {0,1}
  const float* arow = A + (size_t)(rowbase + lm) * KDIM + 2 * hf;
  const float* bptr = Bp + ((size_t)tile * 32 + lane) * 2;

  v8f acc = {};
#pragma unroll 8
  for (int kt = 0; kt < KDIM / 4; ++kt) {
    const v2f a = *(const v2f*)(arow + kt * 4);
    const v2f b = *(const v2f*)(bptr + (size_t)kt * (NTILES * 64));
    acc = __builtin_amdgcn_wmma_f32_16x16x4_f32(false, a, false, b,
                                                (short)0, acc, false, false);
  }

#pragma unroll
  for (int v = 0; v < 8; ++v)
    C[(size_t)(rowbase + v + 8 * hf) * NCOLS + col] = acc[v];
}

// -------------------------------------------------------------------------
// Per-node/head attention scores: el[n,h] = <feat[n,h,:], al[h,:]>, same er.
// -------------------------------------------------------------------------
__global__ void gat_scores(const float* __restrict__ feat,
                           const float* __restrict__ al,
                           const float* __restrict__ ar,
                           float* __restrict__ el, float* __restrict__ er,
                           int D, int ld) {
  const int t = blockIdx.x * blockDim.x + threadIdx.x;
  const int n = t >> 2;
  const int h = t & 3;
  if (n >= N_NODES) return;
  const float* f   = feat + (size_t)n * ld + h * D;
  const float* alp = al + h * D;
  const float* arp = ar + h * D;
  float sl = 0.0f, sr = 0.0f;
  for (int d = 0; d < D; ++d) {
    const float v = f[d];
    sl += v * alp[d];
    sr += v * arp[d];
  }
  el[n * 4 + h] = sl;
  er[n * 4 + h] = sr;
}

// -------------------------------------------------------------------------
// Edge-softmax + aggregation, 256-wide layers. One wave per destination node.
// -------------------------------------------------------------------------
__global__ void gat_agg256(const int* __restrict__ col_ind,
                           const float* __restrict__ feat,
                           const float* __restrict__ el,
                           const float* __restrict__ er,
                           float* __restrict__ out) {
  __shared__ int    s_src[8][16];
  __shared__ float4 s_alpha[8][16];

  const int lane = threadIdx.x & 31;
  const int w    = threadIdx.x >> 5;
  const int node = blockIdx.x * 8 + w;
  const int j    = lane & 15;

  const int src = col_ind[node * DEG + j];
  const float4 elv = *(const float4*)(el + (size_t)src * 4);
  const float4 erv = *(const float4*)(er + (size_t)node * 4);

  float alpha[4];
#pragma unroll
  for (int h = 0; h < 4; ++h) {
    const float x = ((const float*)&elv)[h] + ((const float*)&erv)[h];
    const float e = (x > 0.0f) ? x : NEG_SLOPE * x;
    float m = e;
#pragma unroll
    for (int mask = 8; mask >= 1; mask >>= 1)
      m = fmaxf(m, __shfl_xor(m, mask, 32));
    const float ex = __expf(e - m);
    float ssum = ex;
#pragma unroll
    for (int mask = 8; mask >= 1; mask >>= 1)
      ssum += __shfl_xor(ssum, mask, 32);
    alpha[h] = ex / ssum;
  }
  if (lane < 16) {
    s_src[w][j]   = src;
    s_alpha[w][j] = make_float4(alpha[0], alpha[1], alpha[2], alpha[3]);
  }
  __syncthreads();

  float4 acc0 = make_float4(0.f, 0.f, 0.f, 0.f);
  float4 acc1 = make_float4(0.f, 0.f, 0.f, 0.f);
  const int hsel = lane >> 4;
#pragma unroll
  for (int jj = 0; jj < 16; ++jj) {
    const int    s  = s_src[w][jj];
    const float4 a4 = s_alpha[w][jj];
    const float  a0 = hsel ? a4.y : a4.x;
    const float  a1 = hsel ? a4.w : a4.z;
    const float4* fp = (const float4*)(feat + (size_t)s * 256);
    const float4 f0 = fp[lane];
    const float4 f1 = fp[lane + 32];
    acc0.x += a0 * f0.x; acc0.y += a0 * f0.y; acc0.z += a0 * f0.z; acc0.w += a0 * f0.w;
    acc1.x += a1 * f1.x; acc1.y += a1 * f1.y; acc1.z += a1 * f1.z; acc1.w += a1 * f1.w;
  }
  float4* op = (float4*)(out + (size_t)node * 256);
  op[lane]      = acc0;
  op[lane + 32] = acc1;
}

// -------------------------------------------------------------------------
// Final layer: aggregate 164 used cols of the 176-padded features,
// then mean over the 4 heads.
// -------------------------------------------------------------------------
__global__ void gat_agg_out(const int* __restrict__ col_ind,
                            const float* __restrict__ feat,
                            const float* __restrict__ el,
                            const float* __restrict__ er,
                            float* __restrict__ out) {
  __shared__ int   s_src[8][16];
  __shared__ float s_alpha[8][16][4];
  __shared__ float s_agg[8][L2W];

  const int lane = threadIdx.x & 31;
  const int w    = threadIdx.x >> 5;
  const int node = blockIdx.x * 8 + w;
  const int j    = lane & 15;

  const int src = col_ind[node * DEG + j];
  const float4 elv = *(const float4*)(el + (size_t)src * 4);
  const float4 erv = *(const float4*)(er + (size_t)node * 4);

  float alpha[4];
#pragma unroll
  for (int h = 0; h < 4; ++h) {
    const float x = ((const float*)&elv)[h] + ((const float*)&erv)[h];
    const float e = (x > 0.0f) ? x : NEG_SLOPE * x;
    float m = e;
#pragma unroll
    for (int mask = 8; mask >= 1; mask >>= 1)
      m = fmaxf(m, __shfl_xor(m, mask, 32));
    const float ex = __expf(e - m);
    float ssum = ex;
#pragma unroll
    for (int mask = 8; mask >= 1; mask >>= 1)
      ssum += __shfl_xor(ssum, mask, 32);
    alpha[h] = ex / ssum;
  }
  if (lane < 16) {
    s_src[w][j] = src;
#pragma unroll
    for (int h = 0; h < 4; ++h) s_alpha[w][j][h] = alpha[h];
  }
  __syncthreads();

  float acc[6] = {0.f, 0.f, 0.f, 0.f, 0.f, 0.f};
  int hd[6];
#pragma unroll
  for (int k = 0; k < 6; ++k) {
    const int c = lane + 32 * k;
    hd[k] = (c < L2W) ? (c / OUTD) : 0;
  }
  for (int jj = 0; jj < 16; ++jj) {
    const int    s  = s_src[w][jj];
    const float* fp = feat + (size_t)s * L2WP;   // padded ld
#pragma unroll
    for (int k = 0; k < 6; ++k) {
      const int c = lane + 32 * k;
      if (c < L2W) acc[k] += s_alpha[w][jj][hd[k]] * fp[c];
    }
  }
#pragma unroll
  for (int k = 0; k < 6; ++k) {
    const int c = lane + 32 * k;
    if (c < L2W) s_agg[w][c] = acc[k];
  }
  __syncthreads();

  for (int d = lane; d < OUTD; d += 32) {
    out[(size_t)node * OUTD + d] =
        0.25f * (s_agg[w][d] + s_agg[w][41 + d] + s_agg[w][82 + d] + s_agg[w][123 + d]);
  }
}

// -------------------------------------------------------------------------
extern "C" void kernel_launch(void* const* d_in, const int* in_sizes, int n_in,
                              void* d_out, int out_size, void* d_ws, size_t ws_size,
                              hipStream_t stream) {
  // 0 row_ptr (unused), 1 col_ind, 2 inputs,
  // 3 W0, 4 al0, 5 ar0, 6 W1, 7 al1, 8 ar1, 9 W2, 10 al2, 11 ar2
  const int*   col_ind = (const int*)d_in[1];
  const float* inputs  = (const float*)d_in[2];
  const float* W0  = (const float*)d_in[3];
  const float* al0 = (const float*)d_in[4];
  const float* ar0 = (const float*)d_in[5];
  const float* W1  = (const float*)d_in[6];
  const float* al1 = (const float*)d_in[7];
  const float* ar1 = (const float*)d_in[8];
  const float* W2  = (const float*)d_in[9];
  const float* al2 = (const float*)d_in[10];
  const float* ar2 = (const float*)d_in[11];

  float* ws    = (float*)d_ws;
  float* featA = ws;                                   // [N, 256] (fits in 192MB L2)
  float* hB    = ws + (size_t)N_NODES * 256;           // [N, 256]
  float* el    = hB + (size_t)N_NODES * 256;           // [N, 4]
  float* er    = el + (size_t)N_NODES * 4;             // [N, 4]
  float* Bp    = er + (size_t)N_NODES * 4;             // packed W, <= 256*256 floats
  float* outF  = (float*)d_out;                        // [N, 41]

  const dim3 blk(256);
  const dim3 gGrid256(N_NODES / 16, 2);   // 16 col tiles / 8 waves per block
  const dim3 gGrid176(N_NODES / 16, 2);   // 11 col tiles (guarded)
  const int  scoreBlocks = (N_NODES * 4 + 255) / 256;
  const int  aggBlocks   = N_NODES / 8;
  const int  pack256 = ((KDIM / 4) * 16 * 32 + 255) / 256;  // threads for 16 tiles
  const int  pack176 = ((KDIM / 4) * 11 * 32 + 255) / 256;  // threads for 11 tiles

  // ---- layer 0: inputs -> featA -> hB
  gat_packW <<<pack256, blk, 0, stream>>>(W0, Bp, 256, 16);
  gat_gemm<256><<<gGrid256, blk, 0, stream>>>(inputs, Bp, featA);
  gat_scores<<<scoreBlocks, blk, 0, stream>>>(featA, al0, ar0, el, er, 64, 256);
  gat_agg256<<<aggBlocks,   blk, 0, stream>>>(col_ind, featA, el, er, hB);

  // ---- layer 1: hB -> featA -> hB
  gat_packW <<<pack256, blk, 0, stream>>>(W1, Bp, 256, 16);
  gat_gemm<256><<<gGrid256, blk, 0, stream>>>(hB, Bp, featA);
  gat_scores<<<scoreBlocks, blk, 0, stream>>>(featA, al1, ar1, el, er, 64, 256);
  gat_agg256<<<aggBlocks,   blk, 0, stream>>>(col_ind, featA, el, er, hB);

  // ---- layer 2: hB -> featA(176 padded, 164 used) -> out (head mean)
  gat_packW <<<pack176, blk, 0, stream>>>(W2, Bp, L2W, 11);
  gat_gemm<L2WP><<<gGrid176, blk, 0, stream>>>(hB, Bp, featA);
  gat_scores <<<scoreBlocks, blk, 0, stream>>>(featA, al2, ar2, el, er, OUTD, L2WP);
  gat_agg_out<<<aggBlocks,   blk, 0, stream>>>(col_ind, featA, el, er, outF);
}